// GAT_16836271800954
// MI455X (gfx1250) — compile-verified
//
#include <hip/hip_runtime.h>
#include <hip/hip_bf16.h>

typedef __attribute__((ext_vector_type(2))) float v2f;
typedef __attribute__((ext_vector_type(8))) float v8f;

#define IN_C   128
#define LN     256   // HEADS*OUT_C
#define HEADS  4
#define OUT_C  64
#define EDGE_DIM 32
#define NEG_SLOPE 0.2f

// ---------- monotonic float<->uint mapping for atomic segment-max ----------
__device__ __forceinline__ unsigned fmap(float f) {
    unsigned u = __float_as_uint(f);
    return ((int)u >= 0) ? (u | 0x80000000u) : ~u;
}
__device__ __forceinline__ float funmap(unsigned u) {
    unsigned b = (u & 0x80000000u) ? (u ^ 0x80000000u) : ~u;
    return __uint_as_float(b);
}

// ---------- Kernel 1: x_l = x@W_l + b_l ; x_r = x@W_r + b_r (WMMA fp32) ----
// One block per 16-row tile. 8 waves, each computes 4 of the 32 16x16 tiles
// spanning [W_l | W_r] columns. K=128 swept via 32x v_wmma_f32_16x16x4_f32.
__global__ void __launch_bounds__(256)
k_dual_gemm(const float* __restrict__ x,
            const float* __restrict__ Wl, const float* __restrict__ bl,
            const float* __restrict__ Wr, const float* __restrict__ br,
            float* __restrict__ xl, float* __restrict__ xr, int nrows)
{
    __shared__ float xt[16 * IN_C];   // 8 KB x tile
    const int row0 = blockIdx.x * 16;
    for (int i = threadIdx.x; i < 16 * IN_C; i += 256) {
        int r = i >> 7, c = i & 127;
        int gr = row0 + r;
        xt[i] = (gr < nrows) ? x[(size_t)gr * IN_C + c] : 0.0f;
    }
    __syncthreads();

    const int wave = threadIdx.x >> 5;
    const int lane = threadIdx.x & 31;
    const int m  = lane & 15;             // A row / B col / C col within tile
    const int kq = (lane >> 4) << 1;      // K sub-pair: 0 (lanes 0-15) or 2

    for (int t = 0; t < 4; ++t) {
        const int tt = wave * 4 + t;                     // 0..31
        const float* __restrict__ W    = (tt < 16) ? Wl : Wr;
        const float* __restrict__ bias = (tt < 16) ? bl : br;
        float* __restrict__ out        = (tt < 16) ? xl : xr;
        const int n0 = (tt & 15) * 16;

        v8f acc = {};
        for (int kk = 0; kk < IN_C; kk += 4) {
            v2f a, b;
            // A 16x4 frag (ISA 7.12.2): lane row=m, VGPR0=K kq, VGPR1=K kq+1
            a.x = xt[m * IN_C + kk + kq];
            a.y = xt[m * IN_C + kk + kq + 1];
            // B 4x16 frag: lane col=m, VGPR0=row kq, VGPR1=row kq+1
            b.x = W[(size_t)(kk + kq)     * LN + n0 + m];
            b.y = W[(size_t)(kk + kq + 1) * LN + n0 + m];
            acc = __builtin_amdgcn_wmma_f32_16x16x4_f32(
                      false, a, false, b, (short)0, acc, false, false);
        }
        // C/D layout: VGPR r -> M=r (lanes 0-15) / M=r+8 (lanes 16-31), N=m
        const float bv = bias[n0 + m];
        const int mbase = row0 + ((lane < 16) ? 0 : 8);
        #pragma unroll
        for (int r = 0; r < 8; ++r) {
            int grow = mbase + r;
            if (grow < nrows) out[(size_t)grow * LN + n0 + m] = acc[r] + bv;
        }
    }
}

// ---------- Kernel 2: per-edge attention logits + segment max ---------------
// One wave32 per edge. Lane owns 8 consecutive channels c = lane*8..+7,
// so head h = lane/8 lives in lanes [8h, 8h+7].
__global__ void __launch_bounds__(256)
k_edge_score(const long long* __restrict__ ei, const float* __restrict__ ea,
             const float* __restrict__ We, const float* __restrict__ att,
             const float* __restrict__ xl, const float* __restrict__ xr,
             float* __restrict__ alpha, unsigned* __restrict__ amax_u, int nE)
{
    __shared__ float sWe[EDGE_DIM * LN];  // 32 KB
    __shared__ float satt[LN];
    for (int i = threadIdx.x; i < EDGE_DIM * LN; i += 256) sWe[i] = We[i];
    if (threadIdx.x < LN) satt[threadIdx.x] = att[threadIdx.x];
    __syncthreads();

    const int wave = threadIdx.x >> 5, lane = threadIdx.x & 31;
    const int e = blockIdx.x * 8 + wave;
    if (e >= nE) return;

    const long long src = ei[e];
    const long long dst = ei[(long long)nE + e];

    // e-vector: edge_attr[e] (1x32) @ W_e (32x256), broadcast row via shuffles
    const float eav = ea[(size_t)e * EDGE_DIM + lane];
    const int c0 = lane * 8;
    float acc[8] = {0,0,0,0,0,0,0,0};
    #pragma unroll 4
    for (int k = 0; k < EDGE_DIM; ++k) {
        float v = __shfl(eav, k, 32);
        #pragma unroll
        for (int j = 0; j < 8; ++j) acc[j] += v * sWe[k * LN + c0 + j];
    }

    const float4* pl = (const float4*)(xl + (size_t)src * LN + c0);
    const float4* pr = (const float4*)(xr + (size_t)dst * LN + c0);
    float4 l0 = pl[0], l1 = pl[1], r0 = pr[0], r1 = pr[1];
    float mv[8] = { l0.x + r0.x + acc[0], l0.y + r0.y + acc[1],
                    l0.z + r0.z + acc[2], l0.w + r0.w + acc[3],
                    l1.x + r1.x + acc[4], l1.y + r1.y + acc[5],
                    l1.z + r1.z + acc[6], l1.w + r1.w + acc[7] };
    float s = 0.0f;
    #pragma unroll
    for (int j = 0; j < 8; ++j) {
        float t = mv[j];
        t = (t > 0.0f) ? t : NEG_SLOPE * t;       // LeakyReLU
        s += t * satt[c0 + j];
    }
    // reduce across the 8 lanes of this head
    s += __shfl_xor(s, 1, 32);
    s += __shfl_xor(s, 2, 32);
    s += __shfl_xor(s, 4, 32);
    if ((lane & 7) == 0) {
        int h = lane >> 3;
        alpha[(size_t)e * HEADS + h] = s;
        atomicMax(&amax_u[(size_t)dst * HEADS + h], fmap(s));
    }
}

// ---------- Kernel 3: alpha = exp(alpha - amax[dst]); denom += alpha --------
__global__ void __launch_bounds__(256)
k_exp_denom(const long long* __restrict__ ei, float* __restrict__ alpha,
            const unsigned* __restrict__ amax_u, float* __restrict__ denom, int nE)
{
    int t = blockIdx.x * 256 + threadIdx.x;
    if (t >= nE * HEADS) return;
    int e = t >> 2, h = t & 3;
    long long dst = ei[(long long)nE + e];
    float am = funmap(amax_u[(size_t)dst * HEADS + h]);
    float a = __expf(alpha[t] - am);
    alpha[t] = a;
    atomicAdd(&denom[(size_t)dst * HEADS + h], a);
}

// ---------- Kernel 4: out_acc[dst] += (alpha/denom[dst]) * x_l[src] ---------
__global__ void __launch_bounds__(256)
k_scatter(const long long* __restrict__ ei, const float* __restrict__ alpha,
          const float* __restrict__ denom, const float* __restrict__ xl,
          float* __restrict__ out_acc, int nE)
{
    const int wave = threadIdx.x >> 5, lane = threadIdx.x & 31;
    const int e = blockIdx.x * 8 + wave;
    if (e >= nE) return;
    const long long src = ei[e];
    const long long dst = ei[(long long)nE + e];
    const int h = lane >> 3;
    const float w = alpha[(size_t)e * HEADS + h] / denom[(size_t)dst * HEADS + h];
    const int c0 = lane * 8;
    const float* __restrict__ pl = xl + (size_t)src * LN + c0;
    float* __restrict__ po = out_acc + (size_t)dst * LN + c0;
    #pragma unroll
    for (int j = 0; j < 8; ++j) atomicAdd(&po[j], w * pl[j]);
}

// ---------- Kernel 5: mean over heads + bias --------------------------------
__global__ void __launch_bounds__(256)
k_finalize(const float* __restrict__ out_acc, const float* __restrict__ bias,
           float* __restrict__ out, int nN)
{
    int t = blockIdx.x * 256 + threadIdx.x;
    if (t >= nN * OUT_C) return;
    int i = t >> 6, c = t & 63;
    const float* p = out_acc + (size_t)i * LN + c;
    out[t] = 0.25f * (p[0] + p[OUT_C] + p[2 * OUT_C] + p[3 * OUT_C]) + bias[c];
}

// ---------- Kernel 0: init workspace accumulators ---------------------------
__global__ void __launch_bounds__(256)
k_init(unsigned* __restrict__ amax_u, float* __restrict__ denom,
       float* __restrict__ out_acc, int nN)
{
    int t = blockIdx.x * 256 + threadIdx.x;
    if (t < nN * HEADS) { amax_u[t] = 0u; denom[t] = 0.0f; }  // 0 == fmap(< -inf)
    if (t < nN * LN)    out_acc[t] = 0.0f;
}

extern "C" void kernel_launch(void* const* d_in, const int* in_sizes, int n_in,
                              void* d_out, int out_size, void* d_ws, size_t ws_size,
                              hipStream_t stream) {
    const float*     x    = (const float*)d_in[0];
    const long long* ei   = (const long long*)d_in[1];   // int64 [2, E]
    const float*     ea   = (const float*)d_in[2];
    const float*     Wl   = (const float*)d_in[3];
    const float*     bl   = (const float*)d_in[4];
    const float*     Wr   = (const float*)d_in[5];
    const float*     br   = (const float*)d_in[6];
    const float*     We   = (const float*)d_in[7];
    const float*     att  = (const float*)d_in[8];
    const float*     bias = (const float*)d_in[9];
    float* out = (float*)d_out;

    const int N = in_sizes[0] / IN_C;       // 50000
    const int E = in_sizes[1] / 2;          // 500000

    // workspace layout (fp32): x_l | x_r | alpha | amax_u | denom | out_acc
    float*    xl      = (float*)d_ws;
    float*    xr      = xl + (size_t)N * LN;
    float*    alpha   = xr + (size_t)N * LN;
    unsigned* amax_u  = (unsigned*)(alpha + (size_t)E * HEADS);
    float*    denom   = (float*)(amax_u + (size_t)N * HEADS);
    float*    out_acc = denom + (size_t)N * HEADS;

    k_init<<<(N * LN + 255) / 256, 256, 0, stream>>>(amax_u, denom, out_acc, N);

    k_dual_gemm<<<(N + 15) / 16, 256, 0, stream>>>(x, Wl, bl, Wr, br, xl, xr, N);

    k_edge_score<<<(E + 7) / 8, 256, 0, stream>>>(ei, ea, We, att, xl, xr,
                                                  alpha, amax_u, E);

    k_exp_denom<<<(E * HEADS + 255) / 256, 256, 0, stream>>>(ei, alpha, amax_u,
                                                             denom, E);

    k_scatter<<<(E + 7) / 8, 256, 0, stream>>>(ei, alpha, denom, xl, out_acc, E);

    k_finalize<<<(N * OUT_C + 255) / 256, 256, 0, stream>>>(out_acc, bias, out, N);
}